// StaticMHC_77601469104391
// MI455X (gfx1250) — compile-verified
//
#include <hip/hip_runtime.h>
#include <math.h>

// ---------------------------------------------------------------------------
// StaticMHC fused:  out[t, i*C+c] = sum_j M[i,j] * x[t, j*C+c]
//   M = (1-g)*I + g*Sinkhorn(H_res/tau) + H_post[i]*H_pre[j]   (4x4, folded)
// Memory-bound: 512 MB fp32 traffic, ~0.5 GFLOP -> fp32 end-to-end, streamed
// through V_WMMA_F32_16X16X4_F32 (K = 4 streams, N = 16 channels per tile).
// One wave = one strip (1 token x 128 channels = 8 tiles), three phases:
//   1) 16 loads in one clause (16-deep MLP per wave)
//   2) 8 back-to-back WMMAs (staged loadcnt waits)
//   3) one exec-masked block of 32 stores (single exec toggle)
// ---------------------------------------------------------------------------

#define NSTREAMS 4
#define NC       8192
#define CDIM     2048          // NC / NSTREAMS
#define STRIP_CH 128           // channels per wave strip (8 WMMA tiles)
#define STRIPS_PER_TOKEN 16    // CDIM / STRIP_CH
#define ELEMS_PER_STRIP 512    // NSTREAMS * STRIP_CH
#define TAU      0.05f
#define SK_ITERS 20

typedef float v2f __attribute__((ext_vector_type(2)));
typedef float v8f __attribute__((ext_vector_type(8)));

// ---------------------------------------------------------------------------
// Stage 1: fold Sinkhorn + softmaxes + gate into a single 4x4 matrix M.
// ---------------------------------------------------------------------------
__global__ void mhc_prep_kernel(const float* __restrict__ Hres_logits,
                                const float* __restrict__ Hpre_logits,
                                const float* __restrict__ Hpost_logits,
                                const float* __restrict__ gate,
                                float* __restrict__ Mout)
{
    if (threadIdx.x != 0 || blockIdx.x != 0) return;

    float Z[16];
#pragma unroll
    for (int i = 0; i < 16; ++i) Z[i] = Hres_logits[i] * (1.0f / TAU);

    float u[4] = {0.f, 0.f, 0.f, 0.f};
    float v[4] = {0.f, 0.f, 0.f, 0.f};

    for (int it = 0; it < SK_ITERS; ++it) {
#pragma unroll
        for (int i = 0; i < 4; ++i) {       // u = -lse_j(Z[i,j] + v[j])
            float t0 = Z[i * 4 + 0] + v[0];
            float t1 = Z[i * 4 + 1] + v[1];
            float t2 = Z[i * 4 + 2] + v[2];
            float t3 = Z[i * 4 + 3] + v[3];
            float m = fmaxf(fmaxf(t0, t1), fmaxf(t2, t3));
            float s = expf(t0 - m) + expf(t1 - m) + expf(t2 - m) + expf(t3 - m);
            u[i] = -(m + logf(s));
        }
#pragma unroll
        for (int j = 0; j < 4; ++j) {       // v = -lse_i(Z[i,j] + u[i])
            float t0 = Z[0 * 4 + j] + u[0];
            float t1 = Z[1 * 4 + j] + u[1];
            float t2 = Z[2 * 4 + j] + u[2];
            float t3 = Z[3 * 4 + j] + u[3];
            float m = fmaxf(fmaxf(t0, t1), fmaxf(t2, t3));
            float s = expf(t0 - m) + expf(t1 - m) + expf(t2 - m) + expf(t3 - m);
            v[j] = -(m + logf(s));
        }
    }

    float pre[4], post[4];
    {
        float m = fmaxf(fmaxf(Hpre_logits[0], Hpre_logits[1]),
                        fmaxf(Hpre_logits[2], Hpre_logits[3]));
        float s = 0.f;
#pragma unroll
        for (int j = 0; j < 4; ++j) { pre[j] = expf(Hpre_logits[j] - m); s += pre[j]; }
        float inv = 1.0f / s;
#pragma unroll
        for (int j = 0; j < 4; ++j) pre[j] *= inv;
    }
    {
        float m = fmaxf(fmaxf(Hpost_logits[0], Hpost_logits[1]),
                        fmaxf(Hpost_logits[2], Hpost_logits[3]));
        float s = 0.f;
#pragma unroll
        for (int j = 0; j < 4; ++j) { post[j] = expf(Hpost_logits[j] - m); s += post[j]; }
        float inv = 1.0f / s;
#pragma unroll
        for (int j = 0; j < 4; ++j) post[j] *= inv;
    }

    float g = 1.0f / (1.0f + expf(-gate[0]));

#pragma unroll
    for (int i = 0; i < 4; ++i)
#pragma unroll
        for (int j = 0; j < 4; ++j) {
            float hres = g * expf(Z[i * 4 + j] + u[i] + v[j]);
            if (i == j) hres += (1.0f - g);
            Mout[i * 4 + j] = hres + post[i] * pre[j];
        }
}

// ---------------------------------------------------------------------------
// Stage 2: streaming 4x4 mix via V_WMMA_F32_16X16X4_F32.
//   A (16x4): rows 0..3 = M, rows 4..15 = 0.
//     ISA 7.12.2 slot layout: lanes 0-15 hold K=0 (v0) / K=1 (v1),
//     lanes 16-31 hold K=2 (v0) / K=3 (v1); same slot->K convention used for
//     B, so the K-contraction is layout-permutation safe.
//   B (4x16): B[j, c] = x[token, j*CDIM + cbase + c]  (dense 64B per half).
//   D rows 0..3 = out streams 0..3 at N = lane (lanes 0-15 of d[0..3]).
// ---------------------------------------------------------------------------
__global__ void __launch_bounds__(256)
mhc_mix_wmma_kernel(const float* __restrict__ x,
                    const float* __restrict__ M,
                    float* __restrict__ out,
                    long long nstrips)
{
    const long long strip = (long long)blockIdx.x * (blockDim.x >> 5)
                          + (threadIdx.x >> 5);
    if (strip >= nstrips) return;

    const int lane = threadIdx.x & 31;
    const int half = lane >> 4;     // 0: K={0,1}, 1: K={2,3}
    const int n16  = lane & 15;

    // A operand (strip-invariant): M rows in rows 0..3, zero elsewhere.
    v2f a; a.x = 0.0f; a.y = 0.0f;
    if (n16 < 4) {
        a.x = M[n16 * 4 + 2 * half + 0];   // K = 2*half
        a.y = M[n16 * 4 + 2 * half + 1];   // K = 2*half + 1
    }
    const v8f c0 = {};

    const long long t    = strip >> 4;                         // token
    const int       cb   = (int)(strip & (STRIPS_PER_TOKEN - 1)) * STRIP_CH;
    const long long base = t * (long long)NC + cb;

    const float* __restrict__ px = x + base + (2 * half) * CDIM + n16;
    float* __restrict__       po = out + base + lane;          // lanes 0-15 store

    // Phase 1: issue all 16 loads (one clause, 16-deep in flight).
    v2f b[8];
#pragma unroll
    for (int k = 0; k < 8; ++k) {
        b[k].x = px[16 * k];           // stream 2*half
        b[k].y = px[16 * k + CDIM];    // stream 2*half+1
    }

    // Phase 2: 8 back-to-back WMMAs (distinct D regs; no D->A/B hazards).
    v8f d[8];
#pragma unroll
    for (int k = 0; k < 8; ++k) {
        d[k] = __builtin_amdgcn_wmma_f32_16x16x4_f32(
            /*neg_a=*/false, a, /*neg_b=*/false, b[k],
            /*c_mod=*/(short)0, c0, /*reuse_a=*/false, /*reuse_b=*/false);
    }

    // Phase 3: single exec-masked store block (32 stores, one exec toggle).
    if (lane < 16) {
#pragma unroll
        for (int k = 0; k < 8; ++k) {
            po[16 * k]            = d[k][0];
            po[16 * k +     CDIM] = d[k][1];
            po[16 * k + 2 * CDIM] = d[k][2];
            po[16 * k + 3 * CDIM] = d[k][3];
        }
    }
}

// ---------------------------------------------------------------------------
// Launch
// ---------------------------------------------------------------------------
extern "C" void kernel_launch(void* const* d_in, const int* in_sizes, int n_in,
                              void* d_out, int out_size, void* d_ws, size_t ws_size,
                              hipStream_t stream) {
    const float* x     = (const float*)d_in[0];   // [B, T, 8192] fp32
    const float* Hres  = (const float*)d_in[1];   // [4, 4]
    const float* Hpre  = (const float*)d_in[2];   // [4]
    const float* Hpost = (const float*)d_in[3];   // [4]
    const float* gate  = (const float*)d_in[4];   // [1]
    float*       out   = (float*)d_out;
    float*       Mw    = (float*)d_ws;            // 16-float folded mix matrix

    mhc_prep_kernel<<<1, 32, 0, stream>>>(Hres, Hpre, Hpost, gate, Mw);

    const long long total   = (long long)in_sizes[0];            // B*T*8192
    const long long nstrips = total / ELEMS_PER_STRIP;           // 131072
    const int threads = 256;                                     // 8 wave32
    const int blocks  = (int)((nstrips + 7) / 8);                // 1 strip/wave

    mhc_mix_wmma_kernel<<<blocks, threads, 0, stream>>>(x, Mw, out, nstrips);
}